// WaterMaxProcessor_72859825210034
// MI455X (gfx1250) — compile-verified
//
#include <hip/hip_runtime.h>
#include <stdint.h>

typedef float    f32x4 __attribute__((ext_vector_type(4)));
typedef uint32_t u32x4 __attribute__((ext_vector_type(4)));
typedef uint32_t u32x8 __attribute__((ext_vector_type(8)));

#define NB      128        // batch rows
#define NSEQ    256
#define NV      128000     // vocab
#define NDRAFT  8
#define NFOLD   4
#define WSEED   42u
#define TOPP    0.9f
#define NBINS   2048       // histogram bins over (logit - rowmax) in [-HRANGE, 0]
#define HRANGE  16.0f
#define TILE_F  1024       // floats per async LDS tile (4 KB)
#define THREADS 256        // 8 waves (wave32)
#define CHUNK   1024       // sampling chunk (floats)
#define NCHUNK  (NV / CHUNK)   // 125

static __device__ __forceinline__ uint32_t mix32(uint32_t x) {
  x = (x ^ (x >> 16)) * 0x7FEB352Du;
  x = (x ^ (x >> 15)) * 0x846CA68Bu;
  return x ^ (x >> 16);
}

// CDNA5 async global->LDS copy (ASYNCcnt-tracked, bypasses VGPRs).
static __device__ __forceinline__ void async_tile_b128(uint32_t lds_byte,
                                                       uint32_t voff_byte,
                                                       const void* sbase) {
  asm volatile("global_load_async_to_lds_b128 %0, %1, %2 offset:0"
               :: "v"(lds_byte), "v"(voff_byte), "s"(sbase)
               : "memory");
}
#define WAIT_ASYNC_LE(n) asm volatile("s_wait_asynccnt " #n ::: "memory")

// CDNA5 Tensor Data Mover: 1D contiguous copy of `nelems` f32 from global to
// LDS with a single descriptor (D# group0 in an SGPR quad, group1 in an SGPR
// octet; VADDR2/3 omitted -> NULL, <=2D tensor).  TENSORcnt-tracked.
static __device__ __forceinline__ void tdm_load_1d_f32(uint32_t lds_byte,
                                                       const void* gptr,
                                                       uint32_t nelems) {
  const uint64_t ga = (uint64_t)(uintptr_t)gptr;
  u32x4 g0;
  g0[0] = 1u;                                             // count=1, user D#
  g0[1] = lds_byte;                                       // lds_addr
  g0[2] = (uint32_t)ga;                                   // global_addr[31:0]
  g0[3] = (uint32_t)((ga >> 32) & 0x01FFFFFFu) | (2u << 30); // ga[56:32] | type=2
  u32x8 g1;
  g1[0] = 2u << 16;                                       // data_size = 4 B
  g1[1] = (nelems & 0xFFFFu) << 16;                       // tensor_dim0[15:0]
  g1[2] = (nelems >> 16) | (1u << 16);                    // tensor_dim0[31:16] | tensor_dim1=1
  g1[3] = (nelems & 0xFFFFu) << 16;                       // tile_dim0 = nelems
  g1[4] = 0u;                                             // tile_dim1/2 unused
  g1[5] = nelems;                                         // tensor_dim0_stride
  g1[6] = 0u;
  g1[7] = 0u;
  asm volatile("tensor_load_to_lds %0, %1" :: "s"(g0), "s"(g1) : "memory");
}

// ---------------------------------------------------------------------------
// Kernel 1: per-row max of logits (streaming b128 loads + prefetch).
// ---------------------------------------------------------------------------
__global__ void __launch_bounds__(THREADS)
rowmax_kernel(const float* __restrict__ logits, float* __restrict__ rowmax) {
  const int b = blockIdx.x, tid = threadIdx.x;
  const f32x4* row = (const f32x4*)(logits + (size_t)b * NV);
  float mx = -3.4e38f;
  for (int i = tid; i < NV / 4; i += THREADS) {
    __builtin_prefetch(row + i + 4 * THREADS, 0, 0);      // global_prefetch_b8
    f32x4 v = row[i];
    mx = fmaxf(fmaxf(mx, v.x), fmaxf(v.y, fmaxf(v.z, v.w)));
  }
  for (int o = 16; o > 0; o >>= 1) mx = fmaxf(mx, __shfl_xor(mx, o));
  __shared__ float red[THREADS / 32];
  if ((tid & 31) == 0) red[tid >> 5] = mx;
  __syncthreads();
  if (tid < 32) {
    float v = (tid < THREADS / 32) ? red[tid] : -3.4e38f;
    for (int o = 16; o > 0; o >>= 1) v = fmaxf(v, __shfl_xor(v, o));
    if (tid == 0) rowmax[b] = v;
  }
}

// ---------------------------------------------------------------------------
// Kernel 2: sum-exp + exp-mass histogram via the double-buffered ASYNCcnt
// global->LDS pipeline; wave32 descending scan of bins -> top-p cutoff.
// ---------------------------------------------------------------------------
__global__ void __launch_bounds__(THREADS)
row_stats_kernel(const float* __restrict__ logits,
                 const float* __restrict__ rowmax,
                 float* __restrict__ thresh,
                 float* __restrict__ zmass) {
  __shared__ __align__(16) float tile[2][TILE_F];   // 8 KB double buffer
  __shared__ float hbins[NBINS];                    // 8 KB histogram
  __shared__ float red[THREADS / 32];
  __shared__ float sS;
  const int b = blockIdx.x, tid = threadIdx.x, lane = tid & 31;
  const float m = rowmax[b];
  const float* row = logits + (size_t)b * NV;

  for (int i = tid; i < NBINS; i += THREADS) hbins[i] = 0.f;
  const uint32_t lds0 = (uint32_t)(uintptr_t)&tile[0][0];
  const uint32_t lds1 = (uint32_t)(uintptr_t)&tile[1][0];
  async_tile_b128(lds0 + tid * 16u, tid * 16u, row);  // prime tile 0
  __syncthreads();

  const int ntiles = NV / TILE_F;                   // 125
  float sum = 0.f;
  for (int t = 0; t < ntiles; ++t) {
    if (t + 1 < ntiles) {
      uint32_t dst = (((t + 1) & 1) ? lds1 : lds0) + tid * 16u;
      async_tile_b128(dst, (uint32_t)(t + 1) * (TILE_F * 4u) + tid * 16u, row);
      WAIT_ASYNC_LE(0x1);   // tile t landed (in-order completion)
    } else {
      WAIT_ASYNC_LE(0x0);
    }
    const f32x4 x = *(const f32x4*)&tile[t & 1][tid * 4];
#pragma unroll
    for (int k = 0; k < 4; ++k) {
      float d = x[k] - m;
      float e = __expf(d);
      sum += e;
      int bin = (int)((d + HRANGE) * ((float)NBINS / HRANGE));
      bin = bin < 0 ? 0 : (bin > NBINS - 1 ? NBINS - 1 : bin);
      atomicAdd(&hbins[bin], e);                     // ds_add_f32
    }
  }
  for (int o = 16; o > 0; o >>= 1) sum += __shfl_xor(sum, o);
  if (lane == 0) red[tid >> 5] = sum;
  __syncthreads();
  if (tid < 32) {
    float v = (tid < THREADS / 32) ? red[tid] : 0.f;
    for (int o = 16; o > 0; o >>= 1) v += __shfl_xor(v, o);
    if (tid == 0) sS = v;
  }
  __syncthreads();
  if (tid < 32) {
    const float S = sS, target = TOPP * S;
    float base = 0.f, Z = S;
    int cut = 0;
    bool found = false;
    for (int c = 0; c < NBINS / 32 && !found; ++c) {
      int bin = NBINS - 1 - (c * 32 + lane);
      float incl = hbins[bin];
      for (int o = 1; o < 32; o <<= 1) { float tt = __shfl_up(incl, o); if (lane >= o) incl += tt; }
      float cum = base + incl;
      unsigned long long mk = __ballot(cum >= target);
      if (mk) {
        int w = (int)__ffsll(mk) - 1;
        cut = NBINS - 1 - (c * 32 + w);
        Z = __shfl(cum, w);
        found = true;
      }
      base += __shfl(incl, 31);
    }
    if (tid == 0) {
      thresh[b] = m + ((float)cut * (HRANGE / (float)NBINS) - HRANGE);
      zmass[b]  = Z;
    }
  }
}

// ---------------------------------------------------------------------------
// Kernel 3a: per-chunk kept-mass partial sums, fully parallel (16000 blocks).
// Each block stages its 4 KB chunk with ONE Tensor Data Mover descriptor.
// ---------------------------------------------------------------------------
__global__ void __launch_bounds__(THREADS)
chunk_mass_kernel(const float* __restrict__ logits,
                  const float* __restrict__ rowmax,
                  const float* __restrict__ thresh,
                  float* __restrict__ csum) {
  __shared__ __align__(16) float tile[CHUNK];
  __shared__ float red[THREADS / 32];
  const int b = blockIdx.y, c = blockIdx.x, tid = threadIdx.x, lane = tid & 31;
  const float m = rowmax[b], th = thresh[b];
  const float* src = logits + (size_t)b * NV + (size_t)c * CHUNK;

  if (tid < 32) {                                   // wave 0 drives the TDM
    tdm_load_1d_f32((uint32_t)(uintptr_t)tile, src, CHUNK);
    __builtin_amdgcn_s_wait_tensorcnt(0);           // s_wait_tensorcnt 0
  }
  __syncthreads();                                  // tile visible to all waves

  const f32x4 x = *(const f32x4*)&tile[tid * 4];
  float s = 0.f;
#pragma unroll
  for (int k = 0; k < 4; ++k) s += (x[k] >= th) ? __expf(x[k] - m) : 0.f;
  for (int o = 16; o > 0; o >>= 1) s += __shfl_xor(s, o);
  if (lane == 0) red[tid >> 5] = s;
  __syncthreads();
  if (tid < 32) {
    float v = (tid < THREADS / 32) ? red[tid] : 0.f;
    for (int o = 16; o > 0; o >>= 1) v += __shfl_xor(v, o);
    if (tid == 0) csum[(size_t)b * NCHUNK + c] = v;
  }
}

// ---------------------------------------------------------------------------
// Kernel 3b: one wave per row, but now only a 125-entry chunk scan plus two
// 32-way descent levels per draw (critical path ~30 steps, not 4000).
// argmax(ndtri(u)) == argmax(u) -> hash compare, no inverse normal CDF.
// ---------------------------------------------------------------------------
__global__ void __launch_bounds__(32)
sample_kernel(const float* __restrict__ logits,
              const int* __restrict__ input_ids,
              const float* __restrict__ rowmax,
              const float* __restrict__ thresh,
              const float* __restrict__ csum,
              int* __restrict__ best) {
  __shared__ float pfx[128];                        // inclusive chunk prefix
  const int b = blockIdx.x, lane = threadIdx.x;
  const float m = rowmax[b], th = thresh[b];
  const float* row = logits + (size_t)b * NV;

  float base = 0.f;
  for (int c0 = 0; c0 < 128; c0 += 32) {
    int c = c0 + lane;
    float v = (c < NCHUNK) ? csum[(size_t)b * NCHUNK + c] : 0.f;
    float incl = v;
    for (int o = 1; o < 32; o <<= 1) { float tt = __shfl_up(incl, o); if (lane >= o) incl += tt; }
    pfx[c] = base + incl;
    base += __shfl(incl, 31);
  }
  const float Z = base;                             // exact kept mass
  __syncthreads();

  int cand[NDRAFT];
#pragma unroll
  for (int j = 0; j < NDRAFT; ++j) {
    uint32_t r = mix32(mix32(1u ^ (uint32_t)b * 0x9E3779B9u) ^
                       ((uint32_t)j * 0x85EBCA6Bu + 0x632BE5ABu));
    const float T = (float)(r >> 8) * (1.0f / 16777216.0f) * Z;

    // locate chunk: #chunks whose inclusive prefix <= T
    int cnt = 0;
    for (int c0 = 0; c0 < 128; c0 += 32) {
      int c = c0 + lane;
      cnt += __popcll(__ballot(c < NCHUNK && pfx[c] <= T));
    }
    const int c = cnt < NCHUNK ? cnt : NCHUNK - 1;
    float rem = T - (c > 0 ? pfx[c - 1] : 0.f);
    const float* chk = row + (size_t)c * CHUNK;

    // level 1: each lane sums 32 consecutive elements of the chunk
    float s1 = 0.f;
    const f32x4* cv = (const f32x4*)(chk + lane * 32);
#pragma unroll
    for (int q = 0; q < 8; ++q) {
      f32x4 v = cv[q];
#pragma unroll
      for (int k = 0; k < 4; ++k) s1 += (v[k] >= th) ? __expf(v[k] - m) : 0.f;
    }
    float incl1 = s1;
    for (int o = 1; o < 32; o <<= 1) { float tt = __shfl_up(incl1, o); if (lane >= o) incl1 += tt; }
    unsigned long long mk1 = __ballot(incl1 > rem);
    const int w1 = mk1 ? ((int)__ffsll(mk1) - 1) : 31;
    const float excl1 = __shfl(incl1, w1) - __shfl(s1, w1);
    const float rem2 = rem - excl1;

    // level 2: one element per lane inside the winning 32-span
    float x = chk[w1 * 32 + lane];
    float e = (x >= th) ? __expf(x - m) : 0.f;
    float incl2 = e;
    for (int o = 1; o < 32; o <<= 1) { float tt = __shfl_up(incl2, o); if (lane >= o) incl2 += tt; }
    unsigned long long mk2 = __ballot(incl2 > rem2);
    const int w2 = mk2 ? ((int)__ffsll(mk2) - 1) : 31;

    cand[j] = c * CHUNK + w1 * 32 + w2;             // wave-uniform
  }

  if (lane == 0) {
    uint32_t kt = WSEED;
    const int* ids = input_ids + (size_t)b * NSEQ + (NSEQ - NFOLD);
#pragma unroll
    for (int i = 0; i < NFOLD; ++i) kt = mix32(kt ^ (uint32_t)ids[i]);
    uint32_t hb = 0; int bj = 0;
#pragma unroll
    for (int j = 0; j < NDRAFT; ++j) {
      uint32_t h = mix32(kt ^ mix32((uint32_t)cand[j])) >> 8; // monotone in ndtri
      if (h > hb) { hb = h; bj = j; }
    }
    best[b] = cand[bj];
  }
}

// ---------------------------------------------------------------------------
// Kernel 4: 65.5 MB output fill with nontemporal b128 stores.
// ---------------------------------------------------------------------------
__global__ void __launch_bounds__(THREADS)
fill_kernel(const int* __restrict__ best, float* __restrict__ out) {
  const int b = blockIdx.y;
  const int bt = best[b];
  const int i4 = blockIdx.x * THREADS + threadIdx.x;
  const int col = i4 * 4;
  f32x4 v;
#pragma unroll
  for (int k = 0; k < 4; ++k) v[k] = (col + k == bt) ? 1e5f : 1e-5f;
  __builtin_nontemporal_store(v, (f32x4*)(out + (size_t)b * NV) + i4);
}

// ---------------------------------------------------------------------------
extern "C" void kernel_launch(void* const* d_in, const int* in_sizes, int n_in,
                              void* d_out, int out_size, void* d_ws, size_t ws_size,
                              hipStream_t stream) {
  (void)in_sizes; (void)n_in; (void)out_size; (void)ws_size;
  const int*   input_ids = (const int*)d_in[0];    // [B, SEQ] int32
  const float* logits    = (const float*)d_in[1];  // [B, V]   f32
  float* out = (float*)d_out;                      // [B, V]   f32

  float* rowmax = (float*)d_ws;
  float* thresh = rowmax + NB;
  float* zmass  = thresh + NB;
  int*   best   = (int*)(zmass + NB);
  float* csum   = (float*)(best + NB);             // [NB * NCHUNK]

  rowmax_kernel    <<<NB, THREADS, 0, stream>>>(logits, rowmax);
  row_stats_kernel <<<NB, THREADS, 0, stream>>>(logits, rowmax, thresh, zmass);
  dim3 gc(NCHUNK, NB);                             // 125 x 128 blocks
  chunk_mass_kernel<<<gc, THREADS, 0, stream>>>(logits, rowmax, thresh, csum);
  sample_kernel    <<<NB, 32, 0, stream>>>(logits, input_ids, rowmax, thresh, csum, best);
  dim3 g4(NV / (4 * THREADS), NB);                 // 125 x 128
  fill_kernel      <<<g4, THREADS, 0, stream>>>(best, out);
}